// MultiAgentSystem_22625887715845
// MI455X (gfx1250) — compile-verified
//
#include <hip/hip_runtime.h>
#include <hip/hip_bf16.h>
#include <math.h>

typedef __attribute__((ext_vector_type(2))) float v2f;
typedef __attribute__((ext_vector_type(8))) float v8f;

#define NAGENTS 8192
#define SD 64
#define HD 128
#define CD 32
#define AD 8

// ---------------------------------------------------------------------------
// K1: zero small accumulators in workspace
// ---------------------------------------------------------------------------
__global__ void k_zero(float* __restrict__ p, int n) {
    int i = blockIdx.x * blockDim.x + threadIdx.x;
    if (i < n) p[i] = 0.0f;
}

// ---------------------------------------------------------------------------
// K2: pre1 = states @ W1[:64,:] + b1      (N x 64) x (64 x 128)
// One wave = one 16x16 output tile via v_wmma_f32_16x16x4_f32, K-loop of 16.
// Block = 256 thr = 8 waves = all 8 column tiles of a 16-row stripe.
// ---------------------------------------------------------------------------
__global__ void k_pre1(const float* __restrict__ S, const float* __restrict__ W1,
                       const float* __restrict__ b1, float* __restrict__ pre1) {
    const int lane = threadIdx.x & 31;
    const int wave = threadIdx.x >> 5;          // column tile 0..7
    const int row0 = blockIdx.x * 16;
    const int col0 = wave * 16;
    const int l15  = lane & 15;
    const int koff = (lane < 16) ? 0 : 2;       // f32 WMMA half-wave K split
    const int rb   = (lane < 16) ? 0 : 8;       // C/D row base per half-wave

    v8f c = {};
    for (int k0 = 0; k0 < SD; k0 += 4) {
        v2f a, b;
        a.x = S[(row0 + l15) * SD + k0 + koff];
        a.y = S[(row0 + l15) * SD + k0 + koff + 1];
        b.x = W1[(k0 + koff)     * HD + col0 + l15];
        b.y = W1[(k0 + koff + 1) * HD + col0 + l15];
        c = __builtin_amdgcn_wmma_f32_16x16x4_f32(false, a, false, b,
                                                  (short)0, c, false, false);
    }
    const float bias = b1[col0 + l15];
#pragma unroll
    for (int v = 0; v < 8; ++v)
        pre1[(row0 + rb + v) * HD + col0 + l15] = c[v] + bias;
}

// ---------------------------------------------------------------------------
// K3 (fused): H1 = relu(pre1)@W2 + b2  ->  msgs = l2norm(H1@Wc + bc)
//             msum[j] += sum_rows msgs[:,j]   (for the mean message)
// Block = 256 thr = 8 waves handles one 16-row stripe; H1 stays in LDS.
// ---------------------------------------------------------------------------
__global__ void k_h1_msgs(const float* __restrict__ pre1, const float* __restrict__ W2,
                          const float* __restrict__ b2,  const float* __restrict__ Wc,
                          const float* __restrict__ bc,  float* __restrict__ msum) {
    __shared__ float h1s[16 * 132];   // 16 x 128, padded stride
    __shared__ float ms [16 * 36];    // 16 x 32, padded stride

    const int lane = threadIdx.x & 31;
    const int wave = threadIdx.x >> 5;
    const int row0 = blockIdx.x * 16;
    const int col0 = wave * 16;
    const int l15  = lane & 15;
    const int koff = (lane < 16) ? 0 : 2;
    const int rb   = (lane < 16) ? 0 : 8;

    // H1 tile: K = 128, A = relu(pre1) applied in-register
    v8f c = {};
    for (int k0 = 0; k0 < HD; k0 += 4) {
        v2f a, b;
        a.x = fmaxf(pre1[(row0 + l15) * HD + k0 + koff],     0.0f);
        a.y = fmaxf(pre1[(row0 + l15) * HD + k0 + koff + 1], 0.0f);
        b.x = W2[(k0 + koff)     * HD + col0 + l15];
        b.y = W2[(k0 + koff + 1) * HD + col0 + l15];
        c = __builtin_amdgcn_wmma_f32_16x16x4_f32(false, a, false, b,
                                                  (short)0, c, false, false);
    }
    const float bias = b2[col0 + l15];
#pragma unroll
    for (int v = 0; v < 8; ++v)
        h1s[(rb + v) * 132 + col0 + l15] = c[v] + bias;
    __syncthreads();

    // msgs = H1 @ Wc + bc : two 16x16 tiles (CD = 32), waves 0..1
    if (wave < 2) {
        const int n0 = wave * 16;
        v8f cm = {};
        for (int k0 = 0; k0 < HD; k0 += 4) {
            v2f a, b;
            a.x = h1s[l15 * 132 + k0 + koff];
            a.y = h1s[l15 * 132 + k0 + koff + 1];
            b.x = Wc[(k0 + koff)     * CD + n0 + l15];
            b.y = Wc[(k0 + koff + 1) * CD + n0 + l15];
            cm = __builtin_amdgcn_wmma_f32_16x16x4_f32(false, a, false, b,
                                                       (short)0, cm, false, false);
        }
        const float bcv = bc[n0 + l15];
#pragma unroll
        for (int v = 0; v < 8; ++v)
            ms[(rb + v) * 36 + n0 + l15] = cm[v] + bcv;
    }
    __syncthreads();

    // per-row l2 normalize (F.normalize semantics: x / max(||x||, 1e-12))
    if (threadIdx.x < 16) {
        const int r = threadIdx.x;
        float ss = 0.0f;
        for (int j = 0; j < CD; ++j) { float x = ms[r * 36 + j]; ss += x * x; }
        const float inv = 1.0f / fmaxf(sqrtf(ss), 1e-12f);
        for (int j = 0; j < CD; ++j) ms[r * 36 + j] *= inv;
    }
    __syncthreads();

    // column partial sums -> global accumulator (for mean over N rows)
    if (threadIdx.x < CD) {
        const int j = threadIdx.x;
        float s = 0.0f;
        for (int r = 0; r < 16; ++r) s += ms[r * 36 + j];
        atomicAdd(&msum[j], s);
    }
}

// ---------------------------------------------------------------------------
// K4: cc = (msum / N) @ W1[64:96,:]   (the uniform comm contribution, 128-vec)
// ---------------------------------------------------------------------------
__global__ void k_cc(const float* __restrict__ msum, const float* __restrict__ W1,
                     float* __restrict__ cc) {
    const int j = threadIdx.x;                  // 0..127
    float s = 0.0f;
    for (int k = 0; k < CD; ++k)
        s += (msum[k] * (1.0f / (float)NAGENTS)) * W1[(SD + k) * HD + j];
    cc[j] = s;
}

// ---------------------------------------------------------------------------
// K5: hsum[j] += sum_rows relu(pre1[r,j] + cc[j])    (column sums of h2)
// Block = 128 threads (one per column), 64 rows per block.
// ---------------------------------------------------------------------------
__global__ void k_h2sum(const float* __restrict__ pre1, const float* __restrict__ cc,
                        float* __restrict__ hsum) {
    const int t  = threadIdx.x;                 // column
    const int r0 = blockIdx.x * 64;
    const float ccv = cc[t];
    float s = 0.0f;
    for (int r = 0; r < 64; ++r)
        s += fmaxf(pre1[(r0 + r) * HD + t] + ccv, 0.0f);
    atomicAdd(&hsum[t], s);
}

// ---------------------------------------------------------------------------
// K6: gg = relu( ((hsum/N) @ W2 + b2) @ Wg + bg )    (single shared 128-vec)
// ---------------------------------------------------------------------------
__global__ void k_gg(const float* __restrict__ hsum, const float* __restrict__ W2,
                     const float* __restrict__ b2,   const float* __restrict__ Wg,
                     const float* __restrict__ bg,   float* __restrict__ gg) {
    __shared__ float m[HD];
    __shared__ float g[HD];
    const int j = threadIdx.x;                  // 0..127
    m[j] = hsum[j] * (1.0f / (float)NAGENTS);
    __syncthreads();
    float s = b2[j];
    for (int k = 0; k < HD; ++k) s += m[k] * W2[k * HD + j];
    g[j] = s;
    __syncthreads();
    float s2 = bg[j];
    for (int k = 0; k < HD; ++k) s2 += g[k] * Wg[k * HD + j];
    gg[j] = fmaxf(s2, 0.0f);
}

// ---------------------------------------------------------------------------
// K7: per-agent finalization.
// Hf = l2norm(gg + 0.01*noise); actions=tanh(Hf@Wp+bp); value; role softmax.
// One block (128 threads) per agent.
// ---------------------------------------------------------------------------
__global__ void k_final(const float* __restrict__ noise, const float* __restrict__ gg,
                        const float* __restrict__ Wp, const float* __restrict__ bp,
                        const float* __restrict__ Wv, const float* __restrict__ bv,
                        const float* __restrict__ Wr, const float* __restrict__ br,
                        float* __restrict__ outH, float* __restrict__ outA,
                        float* __restrict__ outV, float* __restrict__ outR) {
    __shared__ float hf[HD];
    __shared__ float red[HD];
    __shared__ float logits[3];
    const int i = blockIdx.x;
    const int t = threadIdx.x;

    const float v = gg[t] + noise[i * HD + t] * 0.01f;
    red[t] = v * v;
    __syncthreads();
    for (int s = 64; s > 0; s >>= 1) {
        if (t < s) red[t] += red[t + s];
        __syncthreads();
    }
    const float inv = 1.0f / fmaxf(sqrtf(red[0]), 1e-12f);
    const float h = v * inv;
    hf[t] = h;
    outH[i * HD + t] = h;
    __syncthreads();

    if (t < AD) {                                // policy head
        float s = bp[t];
        for (int k = 0; k < HD; ++k) s += hf[k] * Wp[k * AD + t];
        outA[i * AD + t] = tanhf(s);
    } else if (t == AD) {                        // value head
        float s = bv[0];
        for (int k = 0; k < HD; ++k) s += hf[k] * Wv[k];
        outV[i] = s;
    } else if (t >= 9 && t < 12) {               // role logits
        float s = br[t - 9];
        for (int k = 0; k < HD; ++k) s += hf[k] * Wr[k * 3 + (t - 9)];
        logits[t - 9] = s;
    }
    __syncthreads();
    if (t < 3) {                                 // 3-way softmax
        const float mx = fmaxf(logits[0], fmaxf(logits[1], logits[2]));
        const float e0 = expf(logits[0] - mx);
        const float e1 = expf(logits[1] - mx);
        const float e2 = expf(logits[2] - mx);
        const float e  = (t == 0) ? e0 : ((t == 1) ? e1 : e2);
        outR[i * 3 + t] = e / (e0 + e1 + e2);
    }
}

// ---------------------------------------------------------------------------
extern "C" void kernel_launch(void* const* d_in, const int* in_sizes, int n_in,
                              void* d_out, int out_size, void* d_ws, size_t ws_size,
                              hipStream_t stream) {
    const float* states = (const float*)d_in[0];
    const float* noise  = (const float*)d_in[1];
    const float* W1     = (const float*)d_in[2];
    const float* b1     = (const float*)d_in[3];
    const float* W2     = (const float*)d_in[4];
    const float* b2     = (const float*)d_in[5];
    const float* Wc     = (const float*)d_in[6];
    const float* bc     = (const float*)d_in[7];
    const float* Wg     = (const float*)d_in[8];
    const float* bg     = (const float*)d_in[9];
    const float* Wp     = (const float*)d_in[10];
    const float* bp     = (const float*)d_in[11];
    const float* Wv     = (const float*)d_in[12];
    const float* bv     = (const float*)d_in[13];
    const float* Wr     = (const float*)d_in[14];
    const float* br     = (const float*)d_in[15];

    float* ws   = (float*)d_ws;
    float* pre1 = ws;                            // N*HD floats
    float* msum = ws + (size_t)NAGENTS * HD;     // 32
    float* hsum = msum + CD;                     // 128
    float* cc   = hsum + HD;                     // 128
    float* gg   = cc + HD;                       // 128

    float* outH = (float*)d_out;                 // N*HD
    float* outA = outH + (size_t)NAGENTS * HD;   // N*AD
    float* outV = outA + (size_t)NAGENTS * AD;   // N
    float* outR = outV + (size_t)NAGENTS;        // N*3

    k_zero   <<<1,    256, 0, stream>>>(msum, CD + HD);
    k_pre1   <<<NAGENTS / 16, 256, 0, stream>>>(states, W1, b1, pre1);
    k_h1_msgs<<<NAGENTS / 16, 256, 0, stream>>>(pre1, W2, b2, Wc, bc, msum);
    k_cc     <<<1,    HD,  0, stream>>>(msum, W1, cc);
    k_h2sum  <<<NAGENTS / 64, HD, 0, stream>>>(pre1, cc, hsum);
    k_gg     <<<1,    HD,  0, stream>>>(hsum, W2, b2, Wg, bg, gg);
    k_final  <<<NAGENTS, HD, 0, stream>>>(noise, gg, Wp, bp, Wv, bv, Wr, br,
                                          outH, outA, outV, outR);
}